// DifferentiableToposAttention_5016521801931
// MI455X (gfx1250) — compile-verified
//
#include <hip/hip_runtime.h>
#include <hip/hip_bf16.h>
#include <stdint.h>

// Problem constants from the reference
#define BB 2
#define SS 1024
#define DD 128
#define KCHUNK 64            // V rows per TDM tile in attn_v

typedef float v2f  __attribute__((ext_vector_type(2)));
typedef float v8f  __attribute__((ext_vector_type(8)));
typedef unsigned int u32x4 __attribute__((ext_vector_type(4)));
typedef int   i32x4 __attribute__((ext_vector_type(4)));
typedef int   i32x8 __attribute__((ext_vector_type(8)));

// ---------------------------------------------------------------------------
// TDM helper: async-load a 2D fp32 tile [rows=KCHUNK x cols=16] from a
// row-major [SS x DD] tensor into LDS (contiguous [KCHUNK][16]).
// D# layout per CDNA5 ISA §8 (Group0 + Group1; 2D => groups 2/3 zero).
// ---------------------------------------------------------------------------
__device__ __forceinline__ void tdm_load_v_tile(const float* gsrc, uint32_t lds_addr)
{
    const uint64_t ga = (uint64_t)(uintptr_t)gsrc;

    u32x4 g0;
    g0[0] = 1u;                                   // count=1 (valid), user mode
    g0[1] = lds_addr;                             // LDS byte address
    g0[2] = (uint32_t)ga;                         // global_addr[31:0]
    g0[3] = (uint32_t)(ga >> 32) | 0x80000000u;   // global_addr[56:32] | type=2

    i32x8 g1;
    g1[0] = 2 << 16;                              // data_size=2 (4 bytes); mask=0
    g1[1] = (DD & 0xFFFF) << 16;                  // tensor_dim0[15:0]=128
    g1[2] = (SS & 0xFFFF) << 16;                  // dim0 hi=0 | tensor_dim1[15:0]=1024
    g1[3] = 16 << 16;                             // dim1 hi=0 | tile_dim0=16
    g1[4] = KCHUNK;                               // tile_dim1=64, tile_dim2=0
    g1[5] = DD;                                   // tensor_dim0_stride = 128 elems
    g1[6] = 0;                                    // stride0 hi | stride1 lo
    g1[7] = 0;

    i32x4 g2 = {0, 0, 0, 0};
    i32x4 g3 = {0, 0, 0, 0};

#if defined(__clang_major__) && (__clang_major__ >= 23)
    i32x8 g4 = {0, 0, 0, 0, 0, 0, 0, 0};
    __builtin_amdgcn_tensor_load_to_lds(g0, g1, g2, g3, g4, 0);
#else
    __builtin_amdgcn_tensor_load_to_lds(g0, g1, g2, g3, 0);
#endif
}

// ---------------------------------------------------------------------------
// Kernel 1: fused QKV projection (exact fp32 WMMA 16x16x4).
// grid = (rowTiles=128, colTiles=8, which=3), block = 32 (one wave).
// ---------------------------------------------------------------------------
__global__ __launch_bounds__(32) void qkv_proj_kernel(
    const float* __restrict__ x,
    const float* __restrict__ Wq, const float* __restrict__ bq,
    const float* __restrict__ Wk, const float* __restrict__ bk,
    const float* __restrict__ Wv, const float* __restrict__ bv,
    float* __restrict__ Qo, float* __restrict__ Ko, float* __restrict__ Vo)
{
    const int lane  = threadIdx.x;
    const int mt    = blockIdx.x;
    const int nt    = blockIdx.y;
    const int which = blockIdx.z;

    const float* W    = (which == 0) ? Wq : (which == 1) ? Wk : Wv;
    const float* bias = (which == 0) ? bq : (which == 1) ? bk : bv;
    float*       Out  = (which == 0) ? Qo : (which == 1) ? Ko : Vo;

    const int m0   = mt * 16;
    const int n0   = nt * 16;
    const int half = lane >> 4;
    const int l16  = lane & 15;

    const float* arow = x + (size_t)(m0 + l16) * DD;
    const float* brow = W + (size_t)(n0 + l16) * DD;

    v8f c = {0.f, 0.f, 0.f, 0.f, 0.f, 0.f, 0.f, 0.f};
    #pragma unroll 4
    for (int kk = 0; kk < DD; kk += 4) {
        v2f a = *(const v2f*)(arow + kk + 2 * half);
        v2f b = *(const v2f*)(brow + kk + 2 * half);
        c = __builtin_amdgcn_wmma_f32_16x16x4_f32(
                false, a, false, b, (short)0, c, false, false);
    }

    #pragma unroll
    for (int r = 0; r < 8; ++r) {
        const int row = m0 + r + 8 * half;
        const int col = n0 + l16;
        float v = c[r] + bias[col];
        if (which < 2)
            v = __builtin_amdgcn_rcpf(1.0f + __expf(-v));   // fast sigmoid
        Out[(size_t)row * DD + col] = v;
    }
}

// ---------------------------------------------------------------------------
// Kernel 2: Lukasiewicz scores + causal fill-0 + softmax (VALU-bound stage).
// grid = (S, B), block = 256 (8 wave32s).
// ---------------------------------------------------------------------------
__global__ __launch_bounds__(256) void scores_softmax_kernel(
    const float* __restrict__ Q, const float* __restrict__ K,
    float* __restrict__ Wout, const int* __restrict__ maskp)
{
    const int b   = blockIdx.y;
    const int q   = blockIdx.x;
    const int tid = threadIdx.x;

    __shared__ float qrow[DD];
    __shared__ float sc[SS];
    __shared__ float red[8];

    const float* Qr = Q + ((size_t)b * SS + q) * DD;
    if (tid < DD) qrow[tid] = Qr[tid];
    __syncthreads();

    const int causal = maskp[0];

    for (int k = tid; k < SS; k += 256) {
        // prefetch next K row for this thread (global_prefetch_b8)
        if (k + 256 < SS)
            __builtin_prefetch(K + ((size_t)b * SS + k + 256) * DD, 0, 1);
        float s;
        if (causal && (k > q)) {
            s = 0.0f;                                    // fill-0, per reference
        } else {
            const float* Kr = K + ((size_t)b * SS + k) * DD;
            float acc = 0.0f;
            #pragma unroll 8
            for (int d = 0; d < DD; d += 4) {
                float4 kv = *(const float4*)(Kr + d);
                acc += fminf(kv.x - qrow[d + 0], 0.0f);
                acc += fminf(kv.y - qrow[d + 1], 0.0f);
                acc += fminf(kv.z - qrow[d + 2], 0.0f);
                acc += fminf(kv.w - qrow[d + 3], 0.0f);
            }
            s = 1.0f + acc * (1.0f / (float)DD);
        }
        sc[k] = s * 10.0f;
    }
    __syncthreads();

    float m = -1e30f;
    for (int k = tid; k < SS; k += 256) m = fmaxf(m, sc[k]);
    #pragma unroll
    for (int off = 16; off > 0; off >>= 1) m = fmaxf(m, __shfl_xor(m, off, 32));
    if ((tid & 31) == 0) red[tid >> 5] = m;
    __syncthreads();
    float bmax = red[0];
    #pragma unroll
    for (int i = 1; i < 8; ++i) bmax = fmaxf(bmax, red[i]);
    __syncthreads();

    float psum = 0.0f;
    for (int k = tid; k < SS; k += 256) {
        float e = __expf(sc[k] - bmax);
        sc[k] = e;
        psum += e;
    }
    #pragma unroll
    for (int off = 16; off > 0; off >>= 1) psum += __shfl_xor(psum, off, 32);
    if ((tid & 31) == 0) red[tid >> 5] = psum;
    __syncthreads();
    float bsum = 0.0f;
    #pragma unroll
    for (int i = 0; i < 8; ++i) bsum += red[i];
    const float inv = __builtin_amdgcn_rcpf(bsum);

    float* wrow = Wout + ((size_t)b * SS + q) * SS;
    for (int k = tid; k < SS; k += 256) wrow[k] = sc[k] * inv;
}

// ---------------------------------------------------------------------------
// Kernel 3: out[b,q,:] = attn[b,q,:] @ V[b]   (M=1024, N=128, K=1024 per batch)
// fp32 WMMA 16x16x4; B-operand (V tile, 512B-strided columns) staged into LDS
// by the Tensor Data Mover, double-buffered, synced with s_wait_tensorcnt.
// grid = (qTiles=64, nTiles=8, B), block = 32 (one wave per 16x16 tile).
// ---------------------------------------------------------------------------
__global__ __launch_bounds__(32) void attn_v_kernel(
    const float* __restrict__ Wt, const float* __restrict__ V,
    float* __restrict__ out)
{
    __shared__ float vbuf[2][KCHUNK * 16];   // 2 x 4KB double buffer

    const int lane = threadIdx.x;
    const int qt   = blockIdx.x;
    const int nt   = blockIdx.y;
    const int b    = blockIdx.z;

    const int half = lane >> 4;
    const int l16  = lane & 15;

    const float* Arow = Wt + ((size_t)b * SS + qt * 16 + l16) * SS;
    const float* Vcol = V + ((size_t)b * SS) * DD + nt * 16;   // tile col base

    const uint32_t lds0 = (uint32_t)(uintptr_t)&vbuf[0][0];
    const uint32_t lds1 = (uint32_t)(uintptr_t)&vbuf[1][0];

    const int NCH = SS / KCHUNK;             // 16 chunks

    // issue chunk 0
    tdm_load_v_tile(Vcol, lds0);

    v8f c = {0.f, 0.f, 0.f, 0.f, 0.f, 0.f, 0.f, 0.f};
    for (int ci = 0; ci < NCH; ++ci) {
        if (ci + 1 < NCH) {
            // kick off next tile while computing this one
            tdm_load_v_tile(Vcol + (size_t)(ci + 1) * KCHUNK * DD,
                            ((ci + 1) & 1) ? lds1 : lds0);
            __builtin_amdgcn_s_wait_tensorcnt(1);   // chunk ci resident
        } else {
            __builtin_amdgcn_s_wait_tensorcnt(0);   // last chunk resident
        }

        const float* lb     = vbuf[ci & 1];
        const float* achunk = Arow + ci * KCHUNK;

        #pragma unroll
        for (int kk = 0; kk < KCHUNK; kk += 4) {
            v2f a = *(const v2f*)(achunk + kk + 2 * half);   // attn[q, k..]
            v2f bm;
            bm.x = lb[(kk + 2 * half + 0) * 16 + l16];       // V[k, n] from LDS
            bm.y = lb[(kk + 2 * half + 1) * 16 + l16];
            c = __builtin_amdgcn_wmma_f32_16x16x4_f32(
                    false, a, false, bm, (short)0, c, false, false);
        }
    }

    #pragma unroll
    for (int r = 0; r < 8; ++r) {
        const int row = qt * 16 + r + 8 * half;
        const int col = nt * 16 + l16;
        out[((size_t)b * SS + row) * DD + col] = c[r];
    }
}

// ---------------------------------------------------------------------------
// Launch: ws layout (floats): Q | K | V (each 2*1024*128) | attn (2*1024*1024)
// total = 2,883,584 floats = 11.5 MB of workspace.
// ---------------------------------------------------------------------------
extern "C" void kernel_launch(void* const* d_in, const int* in_sizes, int n_in,
                              void* d_out, int out_size, void* d_ws, size_t ws_size,
                              hipStream_t stream) {
    const float* x    = (const float*)d_in[0];
    const float* Wq   = (const float*)d_in[1];
    const float* bq   = (const float*)d_in[2];
    const float* Wk   = (const float*)d_in[3];
    const float* bk   = (const float*)d_in[4];
    const float* Wv   = (const float*)d_in[5];
    const float* bv   = (const float*)d_in[6];
    const int*   mask = (const int*)d_in[7];

    float* ws   = (float*)d_ws;
    float* Qb   = ws;
    float* Kb   = ws + (size_t)BB * SS * DD;
    float* Vb   = ws + (size_t)2 * BB * SS * DD;
    float* attn = ws + (size_t)3 * BB * SS * DD;
    float* out  = (float*)d_out;

    qkv_proj_kernel<<<dim3((BB * SS) / 16, DD / 16, 3), 32, 0, stream>>>(
        x, Wq, bq, Wk, bk, Wv, bv, Qb, Kb, Vb);

    scores_softmax_kernel<<<dim3(SS, BB), 256, 0, stream>>>(Qb, Kb, attn, mask);

    attn_v_kernel<<<dim3(SS / 16, DD / 16, BB), 32, 0, stream>>>(attn, Vb, out);
}